// L2RU_45732811768497
// MI455X (gfx1250) — compile-verified
//
#include <hip/hip_runtime.h>
#include <math.h>

// ---------------------------------------------------------------------------
// L2RU on MI455X (gfx1250): f32 WMMA (16x16x4) for all sequence-length GEMMs,
// chunk-parallel linear scan (64 chunks x 32 steps) with A resident in LDS
// (288KB of the 320KB/WGP) loaded by the Tensor Data Mover (TDM) with
// hardware row padding, single-WGP LDS-resident LU/Cholesky for set_param.
// ---------------------------------------------------------------------------

typedef __attribute__((ext_vector_type(2))) float v2f;
typedef __attribute__((ext_vector_type(8))) float v8f;
typedef __attribute__((ext_vector_type(4))) unsigned int u32x4;
typedef __attribute__((ext_vector_type(8))) int i32x8;
typedef __attribute__((ext_vector_type(4))) int i32x4;

#define NMAT   256
#define SEQLEN 2048
#define NBATCH 32
#define CHUNK  32
#define NCHUNK 64   // SEQLEN / CHUNK
#define LDA    260  // padded LDS stride (260 % 64 == 4) to avoid bank conflicts

__device__ inline v8f v8zero() {
  v8f z;
#pragma unroll
  for (int i = 0; i < 8; ++i) z[i] = 0.f;
  return z;
}

// D = A(16x4) * B(4x16) + C   (f32 WMMA)
__device__ inline v8f wmma4(v2f a, v2f b, v8f c) {
  return __builtin_amdgcn_wmma_f32_16x16x4_f32(false, a, false, b, (short)0, c,
                                               false, false);
}

// Fragment loader for NT-GEMM operands. For the A operand: rows are M-rows of
// X. For the B operand of C = X @ W^T (W stored row-major [N,K]): rows are the
// N-rows of W.  Lane l: row = row0+(l&15), k = k0 + 2*(l>>4) .. +1.
__device__ inline v2f frag_ld(const float* base, int ld, int row0, int k0,
                              int lane) {
  int r  = row0 + (lane & 15);
  int kk = k0 + ((lane >> 4) << 1);
  const float* p = base + (size_t)r * (size_t)ld + kk;
  v2f v; v.x = p[0]; v.y = p[1];
  return v;
}

// ---------------------------------------------------------------------------
// Tensor Data Mover: DMA a dense 256x256 f32 matrix from global memory into
// LDS at byte offset lds_off, inserting 4 DWORDs of padding after every
// 256-DWORD row so the LDS image has stride LDA=260 floats (bank-conflict
// free for the WMMA fragment pattern).  D# bitfields per CDNA5 ISA ch.8:
//   group0: [1:0]=count=1, [63:32]=lds_addr, [120:64]=global_addr, type=2
//   group1: data_size=4B, pad_enable, pad_interval=7 (256 DW),
//           pad_amount=3 (4 DW), tensor_dim0/1=256, tile_dim0/1=256,
//           tensor_dim0_stride=256
// Tracked with TENSORcnt -> s_wait_tensorcnt 0.
// ---------------------------------------------------------------------------
__device__ inline void tdm_load_256x256_padded(const float* gsrc,
                                               unsigned lds_off) {
  unsigned long long ga = (unsigned long long)(const void*)gsrc;
  u32x4 g0;
  g0[0] = 1u;                                         // count=1, user mode
  g0[1] = lds_off;                                    // lds_addr (bytes)
  g0[2] = (unsigned)(ga & 0xFFFFFFFFu);               // global_addr[31:0]
  g0[3] = (unsigned)((ga >> 32) & 0x1FFFFFFu)         // global_addr[56:32]
          | (2u << 30);                               // type=2 (image)
  i32x8 g1;
  g1[0] = (2 << 16)       // data_size = 2 -> 4 bytes
        | (1 << 20)       // pad_enable
        | (7 << 22)       // pad_interval = 7 -> every 256 DWORDs
        | (3 << 25);      // pad_amount  = 3 -> 4 DWORDs
  g1[1] = (NMAT & 0xFFFF) << 16;  // tensor_dim0[15:0] in bits [63:48]
  g1[2] = (NMAT << 16);           // tensor_dim1[15:0] in bits [95:80]
  g1[3] = (NMAT << 16);           // tile_dim0 in bits [127:112]
  g1[4] = NMAT;                   // tile_dim1 in bits [143:128]
  g1[5] = NMAT;                   // tensor_dim0_stride[31:0]
  g1[6] = 0;
  g1[7] = 0;
  i32x4 z4; z4[0] = 0; z4[1] = 0; z4[2] = 0; z4[3] = 0;
  i32x8 z8;
#pragma unroll
  for (int i = 0; i < 8; ++i) z8[i] = 0;
  // 6-arg form (clang-23 / therock signature): groups 0..3, extra group, cpol
  __builtin_amdgcn_tensor_load_to_lds(g0, g1, z4, z4, z8, 0);
}

// ===========================================================================
// set_param helper kernels (small, O(256^2..256^3))
// ===========================================================================

// Scatter tril/triu parameter vectors into dense matrices; build I+Sk, I-Sk, I.
__global__ void k_build(const float* x11p, const float* x22p, const float* skp,
                        float* X11f, float* X22f, float* Sk, float* IpSk,
                        float* ImSk, float* Ident) {
  int i = blockIdx.x, j = threadIdx.x;
  int idx = i * NMAT + j;
  float x11 = 0.f, x22 = 0.f, sk = 0.f;
  if (j <= i) {
    int t = i * (i + 1) / 2 + j;
    x11 = x11p[t];
    x22 = x22p[t];
  }
  if (j > i) {
    int t = i * (NMAT - 1) - (i * (i - 1)) / 2 + (j - i - 1);
    sk = skp[t];
  } else if (j < i) {
    int t = j * (NMAT - 1) - (j * (j - 1)) / 2 + (i - j - 1);
    sk = -skp[t];
  }
  float d = (i == j) ? 1.f : 0.f;
  X11f[idx] = x11; X22f[idx] = x22; Sk[idx] = sk;
  IpSk[idx] = d + sk; ImSk[idx] = d - sk; Ident[idx] = d;
}

// Small 256x256 GEMM, one thread per output element.
// mode: 0 = A@B, 1 = A@B^T, 2 = A^T@B.   accum: C += vs C =.
__global__ void k_gemm256(const float* A, const float* B, float* C, int mode,
                          int accum) {
  int i = blockIdx.x, j = threadIdx.x;
  float s = 0.f;
  if (mode == 0) {
    for (int k = 0; k < NMAT; ++k) s += A[i * NMAT + k] * B[k * NMAT + j];
  } else if (mode == 1) {
    for (int k = 0; k < NMAT; ++k) s += A[i * NMAT + k] * B[j * NMAT + k];
  } else {
    for (int k = 0; k < NMAT; ++k) s += A[k * NMAT + i] * B[k * NMAT + j];
  }
  int idx = i * NMAT + j;
  C[idx] = accum ? (C[idx] + s) : s;
}

__global__ void k_copy(const float* s, float* d, int n) {
  int i = blockIdx.x * blockDim.x + threadIdx.x;
  if (i < n) d[i] = s[i];
}

__global__ void k_setdiag_dev(float* M, const float* s) {
  int i = blockIdx.x, j = threadIdx.x;
  M[i * NMAT + j] = (i == j) ? *s : 0.f;
}

__global__ void k_scale_copy(const float* src, float* dst, const float* s) {
  int i = blockIdx.x, j = threadIdx.x;
  dst[i * NMAT + j] = src[i * NMAT + j] * (*s);
}

__global__ void k_compV(const float* Z, float* V, const float* scal,
                        const float* gammap) {
  int i = blockIdx.x, j = threadIdx.x;
  float b = scal[0], g = *gammap;
  V[i * NMAT + j] = b * Z[i * NMAT + j] - ((i == j) ? g * g : 0.f);
}

__global__ void k_transpose(const float* S, float* D) {
  int i = blockIdx.x, j = threadIdx.x;
  D[(size_t)j * NMAT + i] = S[(size_t)i * NMAT + j];
}

__global__ void k_negR(const float* R, float* O) {
  int i = blockIdx.x, j = threadIdx.x;
  O[i * NMAT + j] = -0.5f * (R[i * NMAT + j] + R[j * NMAT + i]) +
                    ((i == j) ? 1e-6f : 0.f);
}

__global__ void k_add(const float* X, const float* Y, float* D) {
  int i = blockIdx.x, j = threadIdx.x;
  D[i * NMAT + j] = X[i * NMAT + j] + Y[i * NMAT + j];
}

// Power iteration for lambda_max(Z) (Z is SPD => 2-norm), then beta scalars.
__global__ void k_powiter(const float* Z, const float* gammap,
                          const float* alphap, float* scal) {
  extern __shared__ float sm[];
  float* Zl  = sm;              // 256*257
  float* q   = sm + 256 * 257;  // 256
  float* y   = q + 256;         // 256
  float* red = y + 256;         // 256
  int tid = threadIdx.x;
  for (int idx = tid; idx < NMAT * NMAT; idx += 256)
    Zl[(idx >> 8) * 257 + (idx & 255)] = Z[idx];
  q[tid] = 1.f;
  __syncthreads();
  float nrm = 1.f;
  for (int it = 0; it < 200; ++it) {
    float s = 0.f;
    const float* zr = Zl + tid * 257;
    for (int k = 0; k < NMAT; ++k) s += zr[k] * q[k];
    y[tid] = s;
    red[tid] = s * s;
    __syncthreads();
    for (int st = 128; st > 0; st >>= 1) {
      if (tid < st) red[tid] += red[tid + st];
      __syncthreads();
    }
    nrm = sqrtf(red[0]);
    __syncthreads();
    q[tid] = y[tid] / nrm;
    __syncthreads();
  }
  if (tid == 0) {
    float g = *gammap, al = *alphap;
    float sig = 1.f / (1.f + expf(-al));
    float beta = g * g * sig / nrm;
    scal[0] = beta;
    scal[1] = sqrtf(beta);
    scal[2] = nrm;
  }
}

// LU with partial pivoting in LDS, then 256-RHS fwd/back substitution:
// solves M X = RHS.  Block = 1024 threads.
__global__ void k_lusolve(const float* M, const float* RHS, float* X) {
  extern __shared__ float sm[];
  float* Am   = sm;                       // 256*257
  float* redv = sm + 256 * 257;           // 256
  int*   redi = (int*)(redv + 256);       // 256
  int*   perm = redi + 256;               // 256
  int tid = threadIdx.x;
  int tr = tid >> 5, tc = tid & 31;
  for (int idx = tid; idx < NMAT * NMAT; idx += 1024)
    Am[(idx >> 8) * 257 + (idx & 255)] = M[idx];
  __syncthreads();
  for (int p = 0; p < NMAT; ++p) {
    if (tid < 256) {
      redv[tid] = (tid >= p) ? fabsf(Am[tid * 257 + p]) : -1.f;
      redi[tid] = tid;
    }
    __syncthreads();
    for (int st = 128; st > 0; st >>= 1) {
      if (tid < st && redv[tid + st] > redv[tid]) {
        redv[tid] = redv[tid + st];
        redi[tid] = redi[tid + st];
      }
      __syncthreads();
    }
    int pr = redi[0];
    if (tid == 0) perm[p] = pr;
    if (pr != p && tid < 256) {
      float t = Am[p * 257 + tid];
      Am[p * 257 + tid] = Am[pr * 257 + tid];
      Am[pr * 257 + tid] = t;
    }
    __syncthreads();
    float pinv = 1.f / Am[p * 257 + p];
    for (int r = p + 1 + tid; r < NMAT; r += 1024) Am[r * 257 + p] *= pinv;
    __syncthreads();
    for (int r = p + 1 + tr; r < NMAT; r += 32) {
      float f = Am[r * 257 + p];
      for (int j = p + 1 + tc; j < NMAT; j += 32)
        Am[r * 257 + j] -= f * Am[p * 257 + j];
    }
    __syncthreads();
  }
  for (int idx = tid; idx < NMAT * NMAT; idx += 1024) X[idx] = RHS[idx];
  __syncthreads();
  if (tid < 256) {
    int c = tid;
    for (int p = 0; p < NMAT; ++p) {
      int pr = perm[p];
      if (pr != p) {
        float t = X[p * NMAT + c];
        X[p * NMAT + c] = X[pr * NMAT + c];
        X[pr * NMAT + c] = t;
      }
    }
    for (int r = 0; r < NMAT; ++r) {           // L (unit diag) solve
      float s = X[r * NMAT + c];
      const float* ar = Am + r * 257;
      for (int k = 0; k < r; ++k) s -= ar[k] * X[k * NMAT + c];
      X[r * NMAT + c] = s;
    }
    for (int r = NMAT - 1; r >= 0; --r) {      // U solve
      float s = X[r * NMAT + c];
      const float* ar = Am + r * 257;
      for (int k = r + 1; k < NMAT; ++k) s -= ar[k] * X[k * NMAT + c];
      X[r * NMAT + c] = s / ar[r];
    }
  }
}

// Cholesky (lower) in LDS. Block = 1024 threads.
__global__ void k_chol(const float* M, float* Lout) {
  extern __shared__ float sm[];
  float* Am = sm;  // 256*257
  int tid = threadIdx.x;
  int tr = tid >> 5, tc = tid & 31;
  for (int idx = tid; idx < NMAT * NMAT; idx += 1024)
    Am[(idx >> 8) * 257 + (idx & 255)] = M[idx];
  __syncthreads();
  for (int p = 0; p < NMAT; ++p) {
    if (tid == 0) Am[p * 257 + p] = sqrtf(Am[p * 257 + p]);
    __syncthreads();
    float dinv = 1.f / Am[p * 257 + p];
    for (int r = p + 1 + tid; r < NMAT; r += 1024) Am[r * 257 + p] *= dinv;
    __syncthreads();
    for (int r = p + 1 + tr; r < NMAT; r += 32) {
      float f = Am[r * 257 + p];
      for (int j = p + 1 + tc; j <= r; j += 32)
        Am[r * 257 + j] -= f * Am[j * 257 + p];
    }
    __syncthreads();
  }
  for (int idx = tid; idx < NMAT * NMAT; idx += 1024) {
    int i = idx >> 8, j = idx & 255;
    Lout[idx] = (j <= i) ? Am[i * 257 + j] : 0.f;
  }
}

// Solve (Lfac^T) X = G  where Lfac is lower triangular (so Lfac^T is upper).
__global__ void k_trisolve_ut(const float* Lfac, const float* G, float* X) {
  extern __shared__ float sm[];
  float* Cl = sm;  // 256*257
  int tid = threadIdx.x;
  for (int idx = tid; idx < NMAT * NMAT; idx += 256)
    Cl[(idx >> 8) * 257 + (idx & 255)] = Lfac[idx];
  __syncthreads();
  int c = tid;
  for (int r = NMAT - 1; r >= 0; --r) {
    float s = G[r * NMAT + c];
    for (int k = r + 1; k < NMAT; ++k) s -= Cl[k * 257 + r] * X[k * NMAT + c];
    X[r * NMAT + c] = s / Cl[r * 257 + r];
  }
}

// ===========================================================================
// Main-phase WMMA kernels
// ===========================================================================

// out[M=65536, 256] = X[M,256] @ W[256,256]^T.  Each wave owns a 16x64 strip.
__global__ void k_gemm_nt_wmma(const float* X, const float* W, float* out) {
  int wave = (blockIdx.x * blockDim.x + threadIdx.x) >> 5;
  int lane = threadIdx.x & 31;
  int row0 = (wave >> 2) << 4;
  int col0 = (wave & 3) << 6;
  v8f acc[4];
#pragma unroll
  for (int j = 0; j < 4; ++j) acc[j] = v8zero();
  for (int k0 = 0; k0 < NMAT; k0 += 4) {
    v2f a = frag_ld(X, NMAT, row0, k0, lane);
#pragma unroll
    for (int j = 0; j < 4; ++j) {
      v2f b = frag_ld(W, NMAT, col0 + (j << 4), k0, lane);
      acc[j] = wmma4(a, b, acc[j]);
    }
  }
#pragma unroll
  for (int j = 0; j < 4; ++j)
#pragma unroll
    for (int i = 0; i < 8; ++i) {
      int m = row0 + ((lane >> 4) << 3) + i;
      int n = col0 + (j << 4) + (lane & 15);
      out[(size_t)m * NMAT + n] = acc[j][i];
    }
}

// states[:,0,:] = x0 ; bases[0] = x0
__global__ void k_state0(const float* x0, float* stateg, float* bases) {
  int m = blockIdx.x, n = threadIdx.x;
  float v = x0[m * NMAT + n];
  stateg[((size_t)m * (SEQLEN + 1)) * NMAT + n] = v;
  bases[m * NMAT + n] = v;
}

// Per-chunk local scan from zero state. A (TDM-loaded, 260KB padded image)
// + x (32x260) in LDS.  Writes l_k into states[:, c*CHUNK+k, :], k=1..CHUNK.
__global__ void k_local_scan(const float* Ag, const float* vbuf,
                             float* stateg) {
  extern __shared__ float smem[];
  float* Al = smem;               // 256 rows, stride LDA (TDM pads rows)
  float* xl = smem + 256 * LDA;   // 32*260
  int tid = threadIdx.x, lane = tid & 31, w = tid >> 5;
  // Wave 0 kicks the Tensor Data Mover for the full A matrix; HW inserts
  // 4 DWORDs of pad after every 256-DWORD row -> LDS stride = LDA floats.
  if (w == 0) {
    if (lane == 0) tdm_load_256x256_padded(Ag, 0u);
    __builtin_amdgcn_s_wait_tensorcnt(0);
  }
  for (int idx = tid; idx < NBATCH * NMAT; idx += 256)
    xl[(idx >> 8) * LDA + (idx & 255)] = 0.f;
  __syncthreads();
  int m0 = (w >> 2) << 4;  // 2 M-tiles (BATCH=32)
  int c0 = (w & 3) << 6;   // 4 waves x 64 cols
  int base = blockIdx.x * CHUNK;
  for (int s = 0; s < CHUNK; ++s) {
    int t = base + s;
    v8f acc[4];
#pragma unroll
    for (int j = 0; j < 4; ++j) acc[j] = v8zero();
    for (int k0 = 0; k0 < NMAT; k0 += 4) {
      v2f a = frag_ld(xl, LDA, m0, k0, lane);
#pragma unroll
      for (int j = 0; j < 4; ++j) {
        v2f b = frag_ld(Al, LDA, c0 + (j << 4), k0, lane);
        acc[j] = wmma4(a, b, acc[j]);
      }
    }
    float nx[4][8];
#pragma unroll
    for (int j = 0; j < 4; ++j)
#pragma unroll
      for (int i = 0; i < 8; ++i) {
        int m = m0 + ((lane >> 4) << 3) + i;
        int n = c0 + (j << 4) + (lane & 15);
        float val = acc[j][i] + vbuf[((size_t)m * SEQLEN + t) * NMAT + n];
        stateg[((size_t)m * (SEQLEN + 1) + (t + 1)) * NMAT + n] = val;
        nx[j][i] = val;
      }
    __syncthreads();  // all reads of xl done
#pragma unroll
    for (int j = 0; j < 4; ++j)
#pragma unroll
      for (int i = 0; i < 8; ++i) {
        int m = m0 + ((lane >> 4) << 3) + i;
        int n = c0 + (j << 4) + (lane & 15);
        xl[m * LDA + n] = nx[j][i];
      }
    __syncthreads();
  }
}

// Sequential base pass: bases[c+1] = bases[c] @ (A^CHUNK)^T + l_L(chunk c),
// with QL = A^CHUNK TDM-loaded into LDS.  Single workgroup, NCHUNK-1 steps.
__global__ void k_base_pass(const float* QL, const float* stateg,
                            float* bases) {
  extern __shared__ float smem[];
  float* Al = smem;
  float* xl = smem + 256 * LDA;
  int tid = threadIdx.x, lane = tid & 31, w = tid >> 5;
  if (w == 0) {
    if (lane == 0) tdm_load_256x256_padded(QL, 0u);
    __builtin_amdgcn_s_wait_tensorcnt(0);
  }
  for (int idx = tid; idx < NBATCH * NMAT; idx += 256)
    xl[(idx >> 8) * LDA + (idx & 255)] = bases[idx];
  __syncthreads();
  int m0 = (w >> 2) << 4;
  int c0 = (w & 3) << 6;
  for (int c = 0; c < NCHUNK - 1; ++c) {
    v8f acc[4];
#pragma unroll
    for (int j = 0; j < 4; ++j) acc[j] = v8zero();
    for (int k0 = 0; k0 < NMAT; k0 += 4) {
      v2f a = frag_ld(xl, LDA, m0, k0, lane);
#pragma unroll
      for (int j = 0; j < 4; ++j) {
        v2f b = frag_ld(Al, LDA, c0 + (j << 4), k0, lane);
        acc[j] = wmma4(a, b, acc[j]);
      }
    }
    float nx[4][8];
#pragma unroll
    for (int j = 0; j < 4; ++j)
#pragma unroll
      for (int i = 0; i < 8; ++i) {
        int m = m0 + ((lane >> 4) << 3) + i;
        int n = c0 + (j << 4) + (lane & 15);
        size_t yoff =
            ((size_t)m * (SEQLEN + 1) + (size_t)(c + 1) * CHUNK) * NMAT + n;
        float val = acc[j][i] + stateg[yoff];
        bases[(size_t)(c + 1) * NBATCH * NMAT + (size_t)m * NMAT + n] = val;
        nx[j][i] = val;
      }
    __syncthreads();
#pragma unroll
    for (int j = 0; j < 4; ++j)
#pragma unroll
      for (int i = 0; i < 8; ++i) {
        int m = m0 + ((lane >> 4) << 3) + i;
        int n = c0 + (j << 4) + (lane & 15);
        xl[m * LDA + n] = nx[j][i];
      }
    __syncthreads();
  }
}

// Parallel fixup: states[:, c*CHUNK+k, :] += bases[c] @ (A^k)^T ; k=1..CHUNK.
__global__ void k_fixup(const float* bases, const float* powm, float* stateg) {
  int bx = blockIdx.x;
  int c = bx >> 5;             // chunk
  int k = (bx & 31) + 1;       // power 1..32
  const float* Xb = bases + (size_t)c * NBATCH * NMAT;
  const float* Wq = powm + (size_t)(k - 1) * NMAT * NMAT;
  int tid = threadIdx.x, lane = tid & 31, w = tid >> 5;
  int m0 = (w >> 2) << 4;
  int c0 = (w & 3) << 6;
  int t = c * CHUNK + k;
  v8f acc[4];
#pragma unroll
  for (int j = 0; j < 4; ++j) acc[j] = v8zero();
  for (int k0 = 0; k0 < NMAT; k0 += 4) {
    v2f a = frag_ld(Xb, NMAT, m0, k0, lane);
#pragma unroll
    for (int j = 0; j < 4; ++j) {
      v2f b = frag_ld(Wq, NMAT, c0 + (j << 4), k0, lane);
      acc[j] = wmma4(a, b, acc[j]);
    }
  }
#pragma unroll
  for (int j = 0; j < 4; ++j)
#pragma unroll
    for (int i = 0; i < 8; ++i) {
      int m = m0 + ((lane >> 4) << 3) + i;
      int n = c0 + (j << 4) + (lane & 15);
      size_t off = ((size_t)m * (SEQLEN + 1) + t) * NMAT + n;
      stateg[off] += acc[j][i];
    }
}

// outputs = states[:, :-1, :] @ C^T + input @ D^T (dual-product WMMA GEMM,
// with per-lane row remap for the (SEQ+1)-strided states tensor).
__global__ void k_final_out(const float* stateg, const float* U,
                            const float* Cm, const float* Dm, float* out) {
  int wave = (blockIdx.x * blockDim.x + threadIdx.x) >> 5;
  int lane = threadIdx.x & 31;
  int row0 = (wave >> 2) << 4;
  int col0 = (wave & 3) << 6;
  int r = row0 + (lane & 15);     // global output row = b*SEQ + t
  int bb = r >> 11;               // batch
  int t = r & (SEQLEN - 1);
  const float* srow = stateg + ((size_t)bb * (SEQLEN + 1) + t) * NMAT;
  const float* urow = U + (size_t)r * NMAT;
  int kshift = (lane >> 4) << 1;
  v8f acc[4];
#pragma unroll
  for (int j = 0; j < 4; ++j) acc[j] = v8zero();
  for (int k0 = 0; k0 < NMAT; k0 += 4) {
    int kk = k0 + kshift;
    v2f a1; a1.x = srow[kk]; a1.y = srow[kk + 1];
    v2f a2; a2.x = urow[kk]; a2.y = urow[kk + 1];
#pragma unroll
    for (int j = 0; j < 4; ++j) {
      v2f bc = frag_ld(Cm, NMAT, col0 + (j << 4), k0, lane);
      v2f bd = frag_ld(Dm, NMAT, col0 + (j << 4), k0, lane);
      acc[j] = wmma4(a1, bc, acc[j]);
      acc[j] = wmma4(a2, bd, acc[j]);
    }
  }
#pragma unroll
  for (int j = 0; j < 4; ++j)
#pragma unroll
    for (int i = 0; i < 8; ++i) {
      int m = row0 + ((lane >> 4) << 3) + i;
      int n = col0 + (j << 4) + (lane & 15);
      out[(size_t)m * NMAT + n] = acc[j][i];
    }
}

// ===========================================================================
// Host-side orchestration
// ===========================================================================
extern "C" void kernel_launch(void* const* d_in, const int* in_sizes, int n_in,
                              void* d_out, int out_size, void* d_ws,
                              size_t ws_size, hipStream_t stream) {
  (void)in_sizes; (void)n_in; (void)out_size; (void)ws_size;

  const float* input  = (const float*)d_in[0];  // [32,2048,256]
  const float* state0 = (const float*)d_in[1];  // [32,256]
  const float* gammap = (const float*)d_in[2];
  const float* alphap = (const float*)d_in[3];
  const float* x11p   = (const float*)d_in[4];
  const float* x22p   = (const float*)d_in[5];
  const float* skp    = (const float*)d_in[6];
  const float* X21    = (const float*)d_in[7];
  const float* Cm     = (const float*)d_in[8];
  const float* Dt     = (const float*)d_in[9];

  float* outp   = (float*)d_out;                              // [32,2048,256]
  float* stateg = outp + (size_t)NBATCH * SEQLEN * NMAT;      // [32,2049,256]

  const size_t M2 = (size_t)NMAT * NMAT;
  float* W = (float*)d_ws;
  float* X11f = W +  0 * M2;  float* X22f   = W +  1 * M2;
  float* Sk   = W +  2 * M2;  float* IpSk   = W +  3 * M2;
  float* ImSk = W +  4 * M2;  float* Ident  = W +  5 * M2;
  float* invP = W +  6 * M2;  float* Qm     = W +  7 * M2;
  float* Z    = W +  8 * M2;  float* H11    = W +  9 * M2;
  float* H12  = W + 10 * M2;  float* Vm     = W + 11 * M2;
  float* Vt   = W + 12 * M2;  float* H12t   = W + 13 * M2;
  float* Smat = W + 14 * M2;  float* Rm     = W + 15 * M2;
  float* negR = W + 16 * M2;  float* CRl    = W + 17 * M2;
  float* T1   = W + 18 * M2;  float* CRHl   = W + 19 * M2;
  float* G    = W + 20 * M2;  float* Xs     = W + 21 * M2;
  float* Amat = W + 22 * M2;  float* Bmat   = W + 23 * M2;
  float* Dmat = W + 24 * M2;
  float* scal  = W + 25 * M2;             // beta, sqrt(beta), lambda
  float* powm  = scal + 64;               // A^1..A^32   (32 * 64K floats)
  float* bases = powm + (size_t)CHUNK * M2;  // [NCHUNK][32][256]

  const unsigned LU_LDS   = (256 * 257 + 3 * 256) * sizeof(float);
  const unsigned CH_LDS   = (256 * 257) * sizeof(float);
  const unsigned PI_LDS   = (256 * 257 + 3 * 256) * sizeof(float);
  const unsigned SCAN_LDS = (256 * LDA + 32 * LDA) * sizeof(float);  // 299520

  // ---- set_param ---------------------------------------------------------
  k_build<<<256, 256, 0, stream>>>(x11p, x22p, skp, X11f, X22f, Sk, IpSk,
                                   ImSk, Ident);
  // Z = X21@X21^T + X22@X22^T + Dt^T@Dt + I
  k_copy<<<256, 256, 0, stream>>>(Ident, Z, NMAT * NMAT);
  k_gemm256<<<256, 256, 0, stream>>>(X21, X21, Z, 1, 1);
  k_gemm256<<<256, 256, 0, stream>>>(X22f, X22f, Z, 1, 1);
  k_gemm256<<<256, 256, 0, stream>>>(Dt, Dt, Z, 2, 1);
  k_powiter<<<1, 256, PI_LDS, stream>>>(Z, gammap, alphap, scal);
  // H11 = X11@X11^T + C^T@C + beta*I
  k_setdiag_dev<<<256, 256, 0, stream>>>(H11, scal);
  k_gemm256<<<256, 256, 0, stream>>>(X11f, X11f, H11, 1, 1);
  k_gemm256<<<256, 256, 0, stream>>>(Cm, Cm, H11, 2, 1);
  // H12 = sqrt(beta)*(X11@X21^T + C^T@Dt)
  k_gemm256<<<256, 256, 0, stream>>>(X11f, X21, H12, 1, 0);
  k_gemm256<<<256, 256, 0, stream>>>(Cm, Dt, H12, 2, 1);
  k_scale_copy<<<256, 256, 0, stream>>>(H12, H12, scal + 1);
  // V = beta*Z - gamma^2*I ; transposes
  k_compV<<<256, 256, 0, stream>>>(Z, Vm, scal, gammap);
  k_transpose<<<256, 256, 0, stream>>>(Vm, Vt);
  k_transpose<<<256, 256, 0, stream>>>(H12, H12t);
  // Qm = (I-Sk) @ inv(I+Sk)
  k_lusolve<<<1, 1024, LU_LDS, stream>>>(IpSk, Ident, invP);
  k_gemm256<<<256, 256, 0, stream>>>(ImSk, invP, Qm, 0, 0);
  // S = solve(V^T, H12^T); R = H12@S; negR
  k_lusolve<<<1, 1024, LU_LDS, stream>>>(Vt, H12t, Smat);
  k_gemm256<<<256, 256, 0, stream>>>(H12, Smat, Rm, 0, 0);
  k_negR<<<256, 256, 0, stream>>>(Rm, negR);
  // Choleskys
  k_chol<<<1, 1024, CH_LDS, stream>>>(negR, CRl);
  k_add<<<256, 256, 0, stream>>>(negR, H11, T1);
  k_chol<<<1, 1024, CH_LDS, stream>>>(T1, CRHl);
  // A = inv(CRH)^T @ Qm @ CR^T   via  CRH^T A = Qm@CR^T
  k_gemm256<<<256, 256, 0, stream>>>(Qm, CRl, G, 1, 0);
  k_trisolve_ut<<<1, 256, CH_LDS, stream>>>(CRHl, G, Amat);
  // B = A @ solve(H12^T, V^T) ; D = sqrt(beta)*Dt
  k_lusolve<<<1, 1024, LU_LDS, stream>>>(H12t, Vt, Xs);
  k_gemm256<<<256, 256, 0, stream>>>(Amat, Xs, Bmat, 0, 0);
  k_scale_copy<<<256, 256, 0, stream>>>(Dt, Dmat, scal + 1);

  // ---- matrix powers A^1..A^CHUNK ---------------------------------------
  k_copy<<<256, 256, 0, stream>>>(Amat, powm, NMAT * NMAT);
  for (int k = 1; k < CHUNK; ++k)
    k_gemm256<<<256, 256, 0, stream>>>(powm + (size_t)(k - 1) * M2, Amat,
                                       powm + (size_t)k * M2, 0, 0);

  // ---- sequence phase (WMMA + TDM) --------------------------------------
  // v = input @ B^T, staged in the outputs region (overwritten at the end)
  k_gemm_nt_wmma<<<2048, 256, 0, stream>>>(input, Bmat, outp);
  k_state0<<<32, 256, 0, stream>>>(state0, stateg, bases);
  k_local_scan<<<NCHUNK, 256, SCAN_LDS, stream>>>(Amat, outp, stateg);
  k_base_pass<<<1, 256, SCAN_LDS, stream>>>(powm + (size_t)(CHUNK - 1) * M2,
                                            stateg, bases);
  k_fixup<<<NCHUNK * CHUNK, 256, 0, stream>>>(bases, powm, stateg);
  // outputs = states[:, :-1, :] @ C^T + input @ D^T
  k_final_out<<<2048, 256, 0, stream>>>(stateg, input, Cm, Dmat, outp);
}